// GConv_1288490189513
// MI455X (gfx1250) — compile-verified
//
#include <hip/hip_runtime.h>
#include <hip/hip_bf16.h>

#define N_NODES 50000
#define N_EDGES 800000
#define DIM 128
#define NLAYERS 3

// LDS W layout: pair-rows of float2 {W[2p][n], W[2p+1][n]}, padded stride to
// de-conflict banks between lane halves (288 floats % 64 banks == 32).
#define SW_STRIDE 144                  // float2 elements per pair-row
#define SW_FLOATS (64 * SW_STRIDE * 2) // 18432 floats = 72 KB

typedef __attribute__((ext_vector_type(2))) float v2f;
typedef __attribute__((ext_vector_type(8))) float v8f;

// ---------------------------------------------------------------------------
// Degree histogram (self-loops folded in later as +1)
// ---------------------------------------------------------------------------
__global__ __launch_bounds__(256) void gcn_degree(const long long* __restrict__ col,
                                                  unsigned* __restrict__ deg,
                                                  int nEdges) {
    int e = blockIdx.x * 256 + threadIdx.x;
    if (e < nEdges) {
        atomicAdd(&deg[(int)col[e]], 1u);
    }
}

__global__ __launch_bounds__(256) void gcn_dinv(const unsigned* __restrict__ deg,
                                                float* __restrict__ dinv,
                                                int n) {
    int i = blockIdx.x * 256 + threadIdx.x;
    if (i < n) {
        dinv[i] = rsqrtf((float)deg[i] + 1.0f);  // +1 for self-loop
    }
}

// ---------------------------------------------------------------------------
// WMMA GEMM: h = z @ W   (fp32, v_wmma_f32_16x16x4_f32)
// Also writes the self-loop contribution: agg = h * dinv(row)^2
// Block = 256 threads = 8 waves; each wave computes a 16-row x 128-col slab.
// W staged in LDS as interleaved K-pairs -> each B frag is one ds_load_b64.
// ---------------------------------------------------------------------------
__global__ __launch_bounds__(256) void gcn_gemm(const float* __restrict__ z,
                                                const float* __restrict__ W,
                                                const float* __restrict__ dinv,
                                                float* __restrict__ h,
                                                float* __restrict__ agg,
                                                int nNodes) {
    __shared__ float sW[SW_FLOATS];  // 72 KB

    const int tid = threadIdx.x;
    // Cooperative load of W with pair-interleaving:
    // sW2[(k>>1)*SW_STRIDE + n] = { W[k][n], W[k+1][n] } for even k.
    for (int c = tid; c < (DIM * DIM) / 4; c += 256) {
        const int k = c >> 5;             // source row 0..127
        const int n = (c & 31) * 4;       // source col base
        const float4 w = ((const float4*)W)[c];
        float* dst = &sW[(((k >> 1) * SW_STRIDE + n) << 1) + (k & 1)];
        dst[0] = w.x;
        dst[2] = w.y;
        dst[4] = w.z;
        dst[6] = w.w;
    }
    __syncthreads();

    const int wave  = tid >> 5;   // 0..7
    const int lane  = tid & 31;
    const int half  = lane >> 4;  // 0: K+0/K+1, 1: K+2/K+3 (32-bit A layout)
    const int l16   = lane & 15;
    const int mBase = blockIdx.x * 128 + wave * 16;

    // A-fragment source row for this lane (clamped; stores are guarded)
    int aRow = mBase + l16;
    if (aRow >= nNodes) aRow = nNodes - 1;
    const float* zrow = z + (size_t)aRow * DIM + half * 2;

    const v2f* sWp = (const v2f*)sW;

    v8f acc[8];
#pragma unroll
    for (int nt = 0; nt < 8; ++nt) acc[nt] = (v8f)0.0f;

    // K loop: 128 in steps of 4 -> 32 iterations, 8 WMMAs each
    for (int kb = 0; kb < DIM; kb += 4) {
        v2f a;
        a.x = zrow[kb];
        a.y = zrow[kb + 1];
        // pair-row holding (kk, kk+1) for this lane half
        const int bRow = (kb >> 1) + half;
#pragma unroll
        for (int nt = 0; nt < 8; ++nt) {
            const v2f b = sWp[bRow * SW_STRIDE + nt * 16 + l16];  // ds_load_b64
            acc[nt] = __builtin_amdgcn_wmma_f32_16x16x4_f32(
                /*neg_a=*/false, a, /*neg_b=*/false, b,
                /*c_mod=*/(short)0, acc[nt],
                /*reuse_a=*/false, /*reuse_b=*/false);
        }
    }

    // C/D layout: VGPR j -> row M = j (lanes 0-15) or M = j+8 (lanes 16-31), N = lane%16
#pragma unroll
    for (int j = 0; j < 8; ++j) {
        const int r = mBase + j + half * 8;
        if (r < nNodes) {
            const float dv = dinv[r];
            const float d2 = dv * dv;  // self-loop norm = dinv^2
            const size_t base = (size_t)r * DIM + l16;
#pragma unroll
            for (int nt = 0; nt < 8; ++nt) {
                const float v = acc[nt][j];
                h[base + nt * 16]   = v;
                agg[base + nt * 16] = v * d2;
            }
        }
    }
}

// ---------------------------------------------------------------------------
// Edge aggregation: one wave per edge, each lane owns 4 channels.
// agg[col] += h[row] * (dinv[row] * dinv[col])
// Gather is a coalesced 512B row read; scatter via global_atomic_add_f32.
// h and agg are L2-resident (25.6 MB each << 192 MB L2).
// ---------------------------------------------------------------------------
__global__ __launch_bounds__(256) void gcn_edge_agg(const long long* __restrict__ rowIdx,
                                                    const long long* __restrict__ colIdx,
                                                    const float* __restrict__ dinv,
                                                    const float* __restrict__ h,
                                                    float* __restrict__ agg,
                                                    int nEdges) {
    const int e = blockIdx.x * 8 + (threadIdx.x >> 5);
    if (e >= nEdges) return;
    const int lane = threadIdx.x & 31;

    const int src = (int)rowIdx[e];
    const int dst = (int)colIdx[e];
    const float nrm = dinv[src] * dinv[dst];

    const float4 v = *(const float4*)(h + (size_t)src * DIM + lane * 4);
    float* a = agg + (size_t)dst * DIM + lane * 4;
    atomicAdd(a + 0, v.x * nrm);
    atomicAdd(a + 1, v.y * nrm);
    atomicAdd(a + 2, v.z * nrm);
    atomicAdd(a + 3, v.w * nrm);
}

// ---------------------------------------------------------------------------
// Finalize: out = PReLU(agg + b), vectorized float4 (N*D divisible by 4)
// ---------------------------------------------------------------------------
__global__ __launch_bounds__(256) void gcn_finalize(const float* __restrict__ agg,
                                                    const float* __restrict__ b,
                                                    const float* __restrict__ alpha,
                                                    float* __restrict__ out,
                                                    int n4) {
    const int i = blockIdx.x * 256 + threadIdx.x;
    if (i >= n4) return;
    const int c = (i & (DIM / 4 - 1)) * 4;  // channel base (D=128 -> 32 float4 groups/row)

    float4 v = ((const float4*)agg)[i];
    float r[4] = {v.x, v.y, v.z, v.w};
#pragma unroll
    for (int k = 0; k < 4; ++k) {
        float t = r[k] + b[c + k];
        r[k] = (t >= 0.0f) ? t : alpha[c + k] * t;
    }
    ((float4*)out)[i] = make_float4(r[0], r[1], r[2], r[3]);
}

// ---------------------------------------------------------------------------
// Host-side launch
// ---------------------------------------------------------------------------
extern "C" void kernel_launch(void* const* d_in, const int* in_sizes, int n_in,
                              void* d_out, int out_size, void* d_ws, size_t ws_size,
                              hipStream_t stream) {
    const float*     x      = (const float*)d_in[0];
    const long long* eidx   = (const long long*)d_in[1];  // [2, E] int64
    const float*     Ws     = (const float*)d_in[2];      // [L, D, D]
    const float*     bs     = (const float*)d_in[3];      // [L, D]
    const float*     alphas = (const float*)d_in[4];      // [L, D]
    float*           out    = (float*)d_out;

    const long long* rowIdx = eidx;            // source
    const long long* colIdx = eidx + N_EDGES;  // target

    // Workspace carve-up (~77 MB total)
    char* ws = (char*)d_ws;
    const size_t ND_BYTES = (size_t)N_NODES * DIM * sizeof(float);
    float*    h    = (float*)(ws);
    float*    agg  = (float*)(ws + ND_BYTES);
    float*    zbuf = (float*)(ws + 2 * ND_BYTES);
    float*    dinv = (float*)(ws + 3 * ND_BYTES);
    unsigned* deg  = (unsigned*)(ws + 3 * ND_BYTES + (size_t)N_NODES * sizeof(float));

    // Degree + symmetric normalization factors
    hipMemsetAsync(deg, 0, (size_t)N_NODES * sizeof(unsigned), stream);
    gcn_degree<<<(N_EDGES + 255) / 256, 256, 0, stream>>>(colIdx, deg, N_EDGES);
    gcn_dinv<<<(N_NODES + 255) / 256, 256, 0, stream>>>(deg, dinv, N_NODES);

    const int gemmGrid = (N_NODES + 127) / 128;      // 391
    const int edgeGrid = (N_EDGES + 7) / 8;          // 100000
    const int n4       = N_NODES * DIM / 4;          // 1,600,000
    const int finGrid  = (n4 + 255) / 256;           // 6250

    const float* zin = x;
    for (int l = 0; l < NLAYERS; ++l) {
        gcn_gemm<<<gemmGrid, 256, 0, stream>>>(zin, Ws + (size_t)l * DIM * DIM, dinv,
                                               h, agg, N_NODES);
        gcn_edge_agg<<<edgeGrid, 256, 0, stream>>>(rowIdx, colIdx, dinv, h, agg, N_EDGES);
        float* zout = (l == NLAYERS - 1) ? out : zbuf;
        gcn_finalize<<<finGrid, 256, 0, stream>>>(agg, bs + l * DIM, alphas + l * DIM,
                                                  zout, n4);
        zin = zbuf;
    }
}